// Net_SDE_45449343926768
// MI455X (gfx1250) — compile-verified
//
#include <hip/hip_runtime.h>
#include <cstddef>

// ---------------------------------------------------------------------------
// Neural-SDE Monte Carlo on MI455X (gfx1250), fp32 WMMA (16x16x4) formulation.
//
// Transposed GEMMs: H^T = W^T @ X^T, so activations live in WMMA accumulators
// with M = feature, N = path (16 paths per wave32). Weights are prepacked into
// LDS in A-fragment order; layer1->layer2 handoff uses cross-half lane
// exchanges (v_permlanex16) instead of an LDS round-trip. Final 64->1 layer is
// a per-lane dot product + one cross-half exchange add.
//
// v2: opaque "zoff" on all LDS indices defeats LICM (v1 spilled hoisted
//     fragments to scratch).
// v3: typed v2f/v4f LDS loads -> ds_load_2addr_b64/b128 with immediate
//     offsets; shuffles moved off the LDS pipe onto v_permlanex16.
// v4: shufflevector bias init so b128 loads land in accumulator blocks.
// v5: ReLU as tied-operand inline asm (v_max_num_f32 %0,0,%0): one op,
//     no canonicalize (fmaxf emits it; fmed3(x,0,inf) gets combined back
//     into maxnum+canonicalize), and no asm-operand register copies.
// ---------------------------------------------------------------------------

typedef float v2f __attribute__((ext_vector_type(2)));
typedef float v4f __attribute__((ext_vector_type(4)));
typedef float v8f __attribute__((ext_vector_type(8)));

#define N_STEPS 360
// LDS layout (float offsets)
#define LDS_L2A 0       // 4 nets * 4096  : layer-2 A-fragments (W1^T)
#define LDS_L1A 16384   // 4 nets * 256   : layer-1 A-fragments (W0^T, bias folded in K=3)
#define LDS_B2  17408   // 4 nets * 64    : layer-2 bias
#define LDS_WO  17664   // 4 nets * 64    : output weights
#define LDS_TOT 17920   // 71680 bytes (CDNA5: up to 320KB per workgroup)

struct NetW { const float *w0, *b0, *w1, *b1, *wo; };
struct Params { NetW net[4]; };   // 0=diffusion, 1=driftV, 2=diffusionV, 3=diffusionV1

__device__ __forceinline__ v8f wmma4(v2f a, v2f b, v8f c) {
  // D(16x16,f32) = A(16x4,f32) x B(4x16,f32) + C
  return __builtin_amdgcn_wmma_f32_16x16x4_f32(false, a, false, b, (short)0, c,
                                               false, false);
}

// Single-instruction in-place ReLU: v_max_num_f32 x, 0, x (tied operand ->
// no canonicalize, no register copies).
__device__ __forceinline__ float relu1(float x) {
  asm("v_max_num_f32 %0, 0, %0" : "+v"(x));
  return x;
}

// Cross-half exchange: every lane gets the value of lane^16 (wave32).
__device__ __forceinline__ float xhalf16(float x) {
#if __has_builtin(__builtin_amdgcn_permlanex16)
  int xi = __float_as_int(x);
  int r = __builtin_amdgcn_permlanex16(xi, xi, 0x76543210, 0xfedcba98,
                                       false, false);
  return __int_as_float(r);
#else
  return __shfl_xor(x, 16, 32);
#endif
}

// Evaluate one MLP (in->64->64->1) for 16 paths (one wave).
// B1 fragment per lane: low half {blo0,blo1} = features K0,K1; high half
// {bhi0,bhi1} = features K2,K3 (K3 input is 1.0 -> multiplies folded bias).
// `zoff` is an opaque runtime zero: added to every LDS index so the loads
// cannot be hoisted out of the 360-step loop.
__device__ __forceinline__ float eval_net(const float* sm, int n, int lane, bool lo,
                                          int zoff,
                                          float blo0, float blo1,
                                          float bhi0, float bhi1) {
  const int h8 = lo ? 0 : 8;
  v2f B1;
  B1.x = lo ? blo0 : bhi0;
  B1.y = lo ? blo1 : bhi1;

  // ---- layer 1: H1^T = relu(W0aug^T @ X^T)  (bias folded into K=3 slot) ----
  const v2f* l1 = (const v2f*)(sm + LDS_L1A + n * 256 + lane * 2 + zoff);
  v8f acc1[4];
#pragma unroll
  for (int mo = 0; mo < 4; ++mo) {
    v2f A = l1[mo * 32];                 // ds_load_b64, imm offset mo*256B
    v8f zc = {0.f, 0.f, 0.f, 0.f, 0.f, 0.f, 0.f, 0.f};
    v8f d = wmma4(A, B1, zc);
#pragma unroll
    for (int r = 0; r < 8; ++r) d[r] = relu1(d[r]);
    acc1[mo] = d;
  }

  // ---- layer 2: H2^T = relu(W1^T @ H1^T + b1_2) ----
  const v4f* b2 = (const v4f*)(sm + LDS_B2 + n * 64 + h8 + zoff);
  v8f acc2[4];
#pragma unroll
  for (int mo = 0; mo < 4; ++mo) {
    v4f ba = b2[mo * 4];                 // ds_load_b128
    v4f bb = b2[mo * 4 + 1];
    acc2[mo] = __builtin_shufflevector(ba, bb, 0, 1, 2, 3, 4, 5, 6, 7);
  }
  const v2f* l2 = (const v2f*)(sm + LDS_L2A + n * 4096 + lane * 2 + zoff);
#pragma unroll
  for (int m = 0; m < 4; ++m) {      // source feature tile of H1^T
#pragma unroll
    for (int j = 0; j < 4; ++j) {    // k-step within the tile: kk = 4m+j
      const int base = (j & 1) * 4;  // j even -> acc regs 0..3, odd -> 4..7
      float a0 = acc1[m][base + 0], a1 = acc1[m][base + 1];
      float a2 = acc1[m][base + 2], a3 = acc1[m][base + 3];
      v2f B2;
      if ((j >> 1) == 0) {
        // low lanes local, high lanes fetch from partner (lane^16)
        float s2 = xhalf16(a2);
        float s3 = xhalf16(a3);
        B2.x = lo ? a0 : s2;
        B2.y = lo ? a1 : s3;
      } else {
        float s0 = xhalf16(a0);
        float s1 = xhalf16(a1);
        B2.x = lo ? s0 : a2;
        B2.y = lo ? s1 : a3;
      }
      const int kk = 4 * m + j;
#pragma unroll
      for (int mo = 0; mo < 4; ++mo) {
        v2f A = l2[(mo * 16 + kk) * 32]; // ds_load_b64, imm offset (mo*16+kk)*256B
        acc2[mo] = wmma4(A, B2, acc2[mo]);
      }
    }
  }

  // ---- output layer: y[path] = relu(H2^T)[:,path] . wout ----
  const v4f* wo = (const v4f*)(sm + LDS_WO + n * 64 + h8 + zoff);
  float part = 0.f;
#pragma unroll
  for (int mo = 0; mo < 4; ++mo) {
    v4f wa = wo[mo * 4];
    v4f wb = wo[mo * 4 + 1];
    part = fmaf(relu1(acc2[mo][0]), wa.x, part);
    part = fmaf(relu1(acc2[mo][1]), wa.y, part);
    part = fmaf(relu1(acc2[mo][2]), wa.z, part);
    part = fmaf(relu1(acc2[mo][3]), wa.w, part);
    part = fmaf(relu1(acc2[mo][4]), wb.x, part);
    part = fmaf(relu1(acc2[mo][5]), wb.y, part);
    part = fmaf(relu1(acc2[mo][6]), wb.z, part);
    part = fmaf(relu1(acc2[mo][7]), wb.w, part);
  }
  return part + xhalf16(part);
}

__global__ __launch_bounds__(256) void sim_kernel(
    Params P, const float* __restrict__ z, const float* __restrict__ z1,
    const float* __restrict__ S0p, const float* __restrict__ V0p,
    const float* __restrict__ ratep, const int* __restrict__ indices,
    float* __restrict__ smat, int MC) {
  __shared__ __align__(16) float sm[LDS_TOT];
  const int tid = threadIdx.x;

  // ---- prepack weights into WMMA A-fragment order (once per block) ----
  for (int n = 0; n < 4; ++n) {
    const float* W1 = P.net[n].w1;  // (64,64) row-major [k][col]
    for (int d = tid; d < 4096; d += 256) {
      int f = d >> 6;                        // fragment = mo*16+kk
      int e = d & 63;
      int lane = e >> 1, q = e & 1;
      int mo = f >> 4, kk = f & 15;
      int kf = 4 * kk + q + 2 * (lane >> 4); // A layout: low lanes K0/K1, high K2/K3
      int col = 16 * mo + (lane & 15);       // output feature (A row)
      sm[LDS_L2A + n * 4096 + d] = W1[kf * 64 + col];
    }
    const float* W0 = P.net[n].w0;  // (din,64) row-major
    const float* B0 = P.net[n].b0;
    const int din = (n == 0) ? 3 : 2;
    {
      int d = tid;  // 256 threads cover 4*64 exactly
      int mo = d >> 6, e = d & 63;
      int lane = e >> 1, q = e & 1;
      int ks = q + 2 * (lane >> 4);
      int col = 16 * mo + (lane & 15);
      float v = 0.f;
      if (ks < din) v = W0[ks * 64 + col];
      else if (ks == 3) v = B0[col];  // bias folded, B1 K=3 input is 1.0
      sm[LDS_L1A + n * 256 + d] = v;
    }
    if (tid < 64) {
      sm[LDS_B2 + n * 64 + tid] = P.net[n].b1[tid];
      sm[LDS_WO + n * 64 + tid] = P.net[n].wo[tid];
    }
  }
  __syncthreads();

  const int lane = tid & 31;
  const int wave = tid >> 5;
  const bool lo = (lane & 16) == 0;
  const int p = lane & 15;
  const int path = (blockIdx.x * 8 + wave) * 16 + p;
  const float* zr = z + (size_t)path * N_STEPS;
  const float* z1r = z1 + (size_t)path * N_STEPS;

  float S = S0p[0];
  float V = V0p[0];
  const float r = ratep[0];
  int im1[12];
#pragma unroll
  for (int k = 0; k < 12; ++k) im1[k] = indices[k] - 1;

  const float h = 1.0f / (float)N_STEPS;
  const float sq = sqrtf(h);

  int zoff = 0;  // opaque zero: defeats LICM of the LDS weight loads

#pragma unroll 1
  for (int i = 0; i < N_STEPS; ++i) {
    asm volatile("" : "+v"(zoff));  // compiler must assume zoff changes
    const float t = (float)i * h;
    const float zt = zr[i];
    const float z1t = z1r[i];
    // diffusion: inputs (t, S, V, [bias]); V-nets: inputs (t, V, 0, [bias])
    const float y0 = eval_net(sm, 0, lane, lo, zoff, t, S, V, 1.0f);
    const float y1 = eval_net(sm, 1, lane, lo, zoff, t, V, 0.0f, 1.0f);  // driftV
    const float y2 = eval_net(sm, 2, lane, lo, zoff, t, V, 0.0f, 1.0f);  // diffusionV
    const float y3 = eval_net(sm, 3, lane, lo, zoff, t, V, 0.0f, 1.0f);  // diffusionV1
    const float dW = sq * zt;
    const float dW1 = sq * z1t;
    float Sn = fmaf(S, r * h, S) + y0 * dW;
    Sn = relu1(Sn);
    const float Vn = V + y1 * h + y2 * dW + y3 * dW1;
    S = Sn;
    V = Vn;
#pragma unroll
    for (int k = 0; k < 12; ++k) {
      if (i == im1[k] && lo) smat[(size_t)k * MC + path] = S;
    }
  }
}

__global__ __launch_bounds__(256) void payoff_kernel(
    const float* __restrict__ smat, const int* __restrict__ indices,
    const float* __restrict__ ratep, float* __restrict__ out, int MC) {
  const float KC[10] = {100.f, 105.f, 110.f, 115.f, 120.f,
                        125.f, 130.f, 135.f, 140.f, 145.f};
  const float KP[10] = {55.f, 60.f, 65.f, 70.f, 75.f,
                        80.f, 85.f, 90.f, 95.f, 100.f};
  const int k = blockIdx.x;  // 0..11
  float acc[40];
#pragma unroll
  for (int t = 0; t < 40; ++t) acc[t] = 0.f;
  const float* row = smat + (size_t)k * (size_t)MC;
  for (int i = threadIdx.x; i < MC; i += 256) {
    const float S = row[i];
#pragma unroll
    for (int s = 0; s < 10; ++s) {
      acc[s]      += relu1(S - KC[s]);  // call_K
      acc[10 + s] += relu1(KP[s] - S);  // put_Kp
      acc[20 + s] += relu1(S - KP[s]);  // call_Kp
      acc[30 + s] += relu1(KC[s] - S);  // put_Kc
    }
  }
  __shared__ float red[256 * 40];
#pragma unroll
  for (int t = 0; t < 40; ++t) red[threadIdx.x * 40 + t] = acc[t];
  __syncthreads();
  if (threadIdx.x < 40) {
    float s = 0.f;
    for (int j = 0; j < 256; ++j) s += red[j * 40 + threadIdx.x];
    const float r = ratep[0];
    const float disc = expf(-r * (float)indices[k] / (float)N_STEPS);
    const int type = threadIdx.x / 10;
    const int st = threadIdx.x % 10;
    out[type * 120 + k * 10 + st] = (s / (float)MC) * disc;
  }
}

extern "C" void kernel_launch(void* const* d_in, const int* in_sizes, int n_in,
                              void* d_out, int out_size, void* d_ws, size_t ws_size,
                              hipStream_t stream) {
  const float* S0 = (const float*)d_in[0];
  const float* V0 = (const float*)d_in[1];
  const float* rate = (const float*)d_in[2];
  const int* indices = (const int*)d_in[4];
  const float* z = (const float*)d_in[5];
  const float* z1 = (const float*)d_in[6];
  const int MC = in_sizes[5] / N_STEPS;

  // JAX pytree dict order (sorted keys): diffusion, diffusionV, diffusionV1,
  // driftV; per net: hidden[ (W0,b0), (W1,b1) ], out(Wout).
  Params P;
  const int s2n[4]  = {0, 2, 3, 1};  // sorted order -> kernel slot
  const int dins[4] = {3, 2, 2, 2};
  if (n_in >= 28) {
    for (int s = 0; s < 4; ++s) {
      const int b = 8 + s * 5;
      NetW& N = P.net[s2n[s]];
      N.w0 = (const float*)d_in[b + 0];
      N.b0 = (const float*)d_in[b + 1];
      N.w1 = (const float*)d_in[b + 2];
      N.b1 = (const float*)d_in[b + 3];
      N.wo = (const float*)d_in[b + 4];
    }
  } else {
    // params delivered as one concatenated flat buffer in leaf order
    const float* pb = (const float*)d_in[8];
    size_t off = 0;
    for (int s = 0; s < 4; ++s) {
      NetW& N = P.net[s2n[s]];
      N.w0 = pb + off; off += (size_t)dins[s] * 64;
      N.b0 = pb + off; off += 64;
      N.w1 = pb + off; off += 4096;
      N.b1 = pb + off; off += 64;
      N.wo = pb + off; off += 64;
    }
  }

  float* smat = (float*)d_ws;  // 12 * MC floats
  (void)ws_size; (void)out_size; (void)n_in;

  sim_kernel<<<dim3(MC / 128), 256, 0, stream>>>(P, z, z1, S0, V0, rate,
                                                 indices, smat, MC);
  payoff_kernel<<<dim3(12), 256, 0, stream>>>(smat, indices, rate,
                                              (float*)d_out, MC);
}